// EnhancedTelomeraseGNN_57939108823123
// MI455X (gfx1250) — compile-verified
//
#include <hip/hip_runtime.h>
#include <hip/hip_bf16.h>

#define NNODES 30000
#define NEDGES 480000
#define FDIM 64
#define FEDIM 16
#define HID 64
#define NHEAD 4
#define HC 256          // NHEAD*HID
#define CIN1 128        // FDIM+HID
#define NGRAPH 64
#define LEAK 0.2f

typedef __attribute__((ext_vector_type(2))) float v2f;
typedef __attribute__((ext_vector_type(8))) float v8f;

// ---------------------------------------------------------------------------
// f32 WMMA GEMM: C[M,NC] = A[M,KK] @ B[KK,NC] (+bias) (+=C if accum)
// NC, KK compile-time so the fully-unrolled K loop uses immediate offsets
// (no per-iteration address arithmetic). M % 16 == 0, NC % 64 == 0, KK % 4 == 0.
// One wave computes a 16x64 tile: 4 v8f accumulators, 4 WMMAs per K-step of 4.
// Tiles enumerated column-major so waves in a block share the B stripe.
// ---------------------------------------------------------------------------
template <int NC, int KK>
__global__ void gemm_wmma_f32(const float* __restrict__ A, const float* __restrict__ B,
                              const float* __restrict__ bias, float* __restrict__ C,
                              int M, int accum) {
    const int wave = blockIdx.x * (blockDim.x >> 5) + (threadIdx.x >> 5);
    const int lane = threadIdx.x & 31;
    const int ntM = M >> 4;
    const int ntN = NC >> 6;
    if (wave >= ntM * ntN) return;
    const int tm = wave % ntM;          // consecutive waves -> consecutive M tiles,
    const int tn = wave / ntM;          // same N stripe (shared B loads)
    const int m0 = tm << 4, n0 = tn << 6;
    const int lr = lane & 15;           // 0..15
    const int hi = lane >> 4;           // 0 or 1

    v8f acc0 = {}, acc1 = {}, acc2 = {}, acc3 = {};
    if (accum) {
        const float* Cp = C + (m0 + (hi << 3)) * NC + n0 + lr;
#pragma unroll
        for (int i = 0; i < 8; ++i) {
            acc0[i] = Cp[i * NC];      acc1[i] = Cp[i * NC + 16];
            acc2[i] = Cp[i * NC + 32]; acc3[i] = Cp[i * NC + 48];
        }
    }

    // A fragment: lane<16 holds (row m0+lr, k+0/1); lane>=16 holds (.., k+2/3)
    const float* Ap = A + (m0 + lr) * KK + (hi << 1);
    // B fragment: lane reads col n0+lr(+16/32/48) at rows k+2*hi and k+2*hi+1
    const float* Bp = B + (hi << 1) * NC + n0 + lr;

#pragma unroll
    for (int it = 0; it < (KK >> 2); ++it) {
        v2f a, b0, b1, b2, b3;
        a.x  = Ap[it * 4];
        a.y  = Ap[it * 4 + 1];
        b0.x = Bp[it * 4 * NC];      b0.y = Bp[it * 4 * NC + NC];
        b1.x = Bp[it * 4 * NC + 16]; b1.y = Bp[it * 4 * NC + NC + 16];
        b2.x = Bp[it * 4 * NC + 32]; b2.y = Bp[it * 4 * NC + NC + 32];
        b3.x = Bp[it * 4 * NC + 48]; b3.y = Bp[it * 4 * NC + NC + 48];
        acc0 = __builtin_amdgcn_wmma_f32_16x16x4_f32(false, a, false, b0, (short)0, acc0, false, false);
        acc1 = __builtin_amdgcn_wmma_f32_16x16x4_f32(false, a, false, b1, (short)0, acc1, false, false);
        acc2 = __builtin_amdgcn_wmma_f32_16x16x4_f32(false, a, false, b2, (short)0, acc2, false, false);
        acc3 = __builtin_amdgcn_wmma_f32_16x16x4_f32(false, a, false, b3, (short)0, acc3, false, false);
    }

    float bv0 = 0.f, bv1 = 0.f, bv2 = 0.f, bv3 = 0.f;
    if (bias) {
        bv0 = bias[n0 + lr];      bv1 = bias[n0 + lr + 16];
        bv2 = bias[n0 + lr + 32]; bv3 = bias[n0 + lr + 48];
    }
    float* Cp = C + (m0 + (hi << 3)) * NC + n0 + lr;
#pragma unroll
    for (int i = 0; i < 8; ++i) {
        Cp[i * NC]      = acc0[i] + bv0;
        Cp[i * NC + 16] = acc1[i] + bv1;
        Cp[i * NC + 32] = acc2[i] + bv2;
        Cp[i * NC + 48] = acc3[i] + bv3;
    }
}

// ---------------------------------------------------------------------------
// Helpers
// ---------------------------------------------------------------------------
__device__ __forceinline__ float atomicMaxFloat(float* addr, float val) {
    if (val >= 0.f)
        return __int_as_float(atomicMax((int*)addr, __float_as_int(val)));
    else
        return __uint_as_float(atomicMin((unsigned int*)addr, __float_as_uint(val)));
}

__device__ __forceinline__ float lrelu(float x) { return x >= 0.f ? x : LEAK * x; }
__device__ __forceinline__ float elu1(float x)  { return x > 0.f ? x : expf(x) - 1.f; }

__global__ void build_h0_kernel(const float* __restrict__ x, const float* __restrict__ node_emb,
                                const int* __restrict__ node_type, float* __restrict__ h0) {
    int t = blockIdx.x * blockDim.x + threadIdx.x;
    if (t >= NNODES * CIN1) return;
    int n = t >> 7, c = t & 127;
    h0[t] = (c < FDIM) ? x[n * FDIM + c] : node_emb[node_type[n] * HID + (c - FDIM)];
}

// M[k,h] = sum_c leW[k, h*64+c] * aeW[h,c]   (M: [64,4])
__global__ void computeM_kernel(const float* __restrict__ leW, const float* __restrict__ aeW,
                                float* __restrict__ Mb) {
    int tid = threadIdx.x;                 // 256 threads
    int k = tid >> 2, h = tid & 3;
    float acc = 0.f;
    for (int c = 0; c < HID; ++c) acc += leW[k * HC + h * HID + c] * aeW[h * HID + c];
    Mb[k * 4 + h] = acc;
}

// P = edge_W @ M ([16,4]), q = edge_b @ M ([4])
__global__ void computePq_kernel(const float* __restrict__ eW, const float* __restrict__ eb,
                                 const float* __restrict__ Mb, float* __restrict__ P,
                                 float* __restrict__ q) {
    int tid = threadIdx.x;                 // 128 threads
    if (tid < 64) {
        int k = tid >> 2, h = tid & 3;
        float acc = 0.f;
        for (int j = 0; j < HID; ++j) acc += eW[k * HID + j] * Mb[j * 4 + h];
        P[k * 4 + h] = acc;
    } else if (tid < 68) {
        int h = tid - 64;
        float acc = 0.f;
        for (int j = 0; j < HID; ++j) acc += eb[j] * Mb[j * 4 + h];
        q[h] = acc;
    }
}

// column means of edge_attr ([E,16]) -> eamean[16]
__global__ void colmean16_kernel(const float* __restrict__ ea, float* __restrict__ eamean) {
    __shared__ float sm[256];
    int col = blockIdx.x;
    float acc = 0.f;
    for (int e = threadIdx.x; e < NEDGES; e += blockDim.x) acc += ea[e * FEDIM + col];
    sm[threadIdx.x] = acc;
    __syncthreads();
    for (int s = 128; s > 0; s >>= 1) {
        if (threadIdx.x < s) sm[threadIdx.x] += sm[threadIdx.x + s];
        __syncthreads();
    }
    if (threadIdx.x == 0) eamean[col] = sm[0] / (float)NEDGES;
}

__global__ void aeloop_kernel(const float* __restrict__ eamean, const float* __restrict__ P,
                              const float* __restrict__ q, float* __restrict__ aeloop) {
    int h = threadIdx.x;
    if (h >= NHEAD) return;
    float acc = q[h];
    for (int k = 0; k < FEDIM; ++k) acc += eamean[k] * P[k * 4 + h];
    aeloop[h] = acc;
}

// a_s[n,h], a_d[n,h] from hbig [N,256]
__global__ void asad_kernel(const float* __restrict__ hbig, const float* __restrict__ asW,
                            const float* __restrict__ adW, float* __restrict__ as_,
                            float* __restrict__ ad_) {
    int t = blockIdx.x * blockDim.x + threadIdx.x;
    if (t >= NNODES * NHEAD) return;
    int n = t >> 2, h = t & 3;
    const float* hp = hbig + n * HC + h * HID;
    float a = 0.f, b = 0.f;
    for (int c = 0; c < HID; ++c) { a += hp[c] * asW[h * HID + c]; b += hp[c] * adW[h * HID + c]; }
    as_[t] = a; ad_[t] = b;
}

// a_e[e,h] = edge_attr[e,:] @ P[:,h] + q[h]
__global__ void edge_ae_kernel(const float* __restrict__ ea, const float* __restrict__ P,
                               const float* __restrict__ q, float* __restrict__ ae) {
    int t = blockIdx.x * blockDim.x + threadIdx.x;
    if (t >= NEDGES * NHEAD) return;
    int e = t >> 2, h = t & 3;
    float acc = q[h];
    const float* ep = ea + e * FEDIM;
#pragma unroll
    for (int k = 0; k < FEDIM; ++k) acc += ep[k] * P[k * 4 + h];
    ae[t] = acc;
}

// self-loop score per node, initialize running max
__global__ void node_init_kernel(const float* __restrict__ as_, const float* __restrict__ ad_,
                                 const float* __restrict__ aeloop, float* __restrict__ sself,
                                 float* __restrict__ mrow) {
    int t = blockIdx.x * blockDim.x + threadIdx.x;
    if (t >= NNODES * NHEAD) return;
    float s = lrelu(as_[t] + ad_[t] + aeloop[t & 3]);
    sself[t] = s;
    mrow[t] = s;
}

__global__ void edge_max_kernel(const int* __restrict__ src, const int* __restrict__ dst,
                                const float* __restrict__ as_, const float* __restrict__ ad_,
                                const float* __restrict__ ae, float* __restrict__ mrow) {
    int e = blockIdx.x * blockDim.x + threadIdx.x;
    if (e >= NEDGES) return;
    int s = src[e], d = dst[e];
#pragma unroll
    for (int h = 0; h < NHEAD; ++h) {
        float sc = lrelu(as_[s * 4 + h] + ad_[d * 4 + h] + ae[e * 4 + h]);
        atomicMaxFloat(&mrow[d * 4 + h], sc);
    }
}

__global__ void den_init_kernel(const float* __restrict__ sself, const float* __restrict__ mrow,
                                float* __restrict__ den) {
    int t = blockIdx.x * blockDim.x + threadIdx.x;
    if (t >= NNODES * NHEAD) return;
    den[t] = expf(sself[t] - mrow[t]);
}

__global__ void edge_den_kernel(const int* __restrict__ src, const int* __restrict__ dst,
                                const float* __restrict__ as_, const float* __restrict__ ad_,
                                const float* __restrict__ ae, const float* __restrict__ mrow,
                                float* __restrict__ den) {
    int e = blockIdx.x * blockDim.x + threadIdx.x;
    if (e >= NEDGES) return;
    int s = src[e], d = dst[e];
#pragma unroll
    for (int h = 0; h < NHEAD; ++h) {
        float sc = lrelu(as_[s * 4 + h] + ad_[d * 4 + h] + ae[e * 4 + h]);
        atomicAdd(&den[d * 4 + h], expf(sc - mrow[d * 4 + h]));
    }
}

// self-loop contribution: initializes out[n,c] (plain store)
__global__ void gat_self_agg_kernel(const float* __restrict__ sself, const float* __restrict__ mrow,
                                    const float* __restrict__ den, const float* __restrict__ hbig,
                                    float* __restrict__ out) {
    int n = blockIdx.x * (blockDim.x >> 5) + (threadIdx.x >> 5);
    int lane = threadIdx.x & 31;
    if (n >= NNODES) return;
    float w[NHEAD];
#pragma unroll
    for (int h = 0; h < NHEAD; ++h)
        w[h] = 0.25f * expf(sself[n * 4 + h] - mrow[n * 4 + h]) / den[n * 4 + h];
    const float* hp = hbig + n * HC;
#pragma unroll
    for (int j = 0; j < 2; ++j) {
        int c = lane + 32 * j;
        out[n * HID + c] =
            w[0] * hp[c] + w[1] * hp[64 + c] + w[2] * hp[128 + c] + w[3] * hp[192 + c];
    }
}

// warp per edge: weighted gather of h[src] (head-mean folded), scatter-add to out[dst]
__global__ void edge_agg_kernel(const int* __restrict__ src, const int* __restrict__ dst,
                                const float* __restrict__ as_, const float* __restrict__ ad_,
                                const float* __restrict__ ae, const float* __restrict__ mrow,
                                const float* __restrict__ den, const float* __restrict__ hbig,
                                float* __restrict__ out) {
    int e = blockIdx.x * (blockDim.x >> 5) + (threadIdx.x >> 5);
    int lane = threadIdx.x & 31;
    if (e >= NEDGES) return;
    int s = src[e], d = dst[e];
    float w[NHEAD];
#pragma unroll
    for (int h = 0; h < NHEAD; ++h) {
        float sc = lrelu(as_[s * 4 + h] + ad_[d * 4 + h] + ae[e * 4 + h]);
        w[h] = 0.25f * expf(sc - mrow[d * 4 + h]) / den[d * 4 + h];
    }
    const float* hp = hbig + s * HC;
#pragma unroll
    for (int j = 0; j < 2; ++j) {
        int c = lane + 32 * j;
        float v = w[0] * hp[c] + w[1] * hp[64 + c] + w[2] * hp[128 + c] + w[3] * hp[192 + c];
        atomicAdd(&out[d * HID + c], v);
    }
}

__global__ void bias_elu_kernel(float* __restrict__ h, const float* __restrict__ b) {
    int t = blockIdx.x * blockDim.x + threadIdx.x;
    if (t >= NNODES * HID) return;
    h[t] = elu1(h[t] + b[t & 63]);
}

__global__ void elu_kernel(float* __restrict__ h, int total) {
    int t = blockIdx.x * blockDim.x + threadIdx.x;
    if (t >= total) return;
    h[t] = elu1(h[t]);
}

__global__ void sage_deg_kernel(const int* __restrict__ dst, float* __restrict__ deg) {
    int e = blockIdx.x * blockDim.x + threadIdx.x;
    if (e >= NEDGES) return;
    atomicAdd(&deg[dst[e]], 1.f);
}

__global__ void sage_agg_kernel(const int* __restrict__ src, const int* __restrict__ dst,
                                const float* __restrict__ h2, float* __restrict__ agg) {
    int e = blockIdx.x * (blockDim.x >> 5) + (threadIdx.x >> 5);
    int lane = threadIdx.x & 31;
    if (e >= NEDGES) return;
    int s = src[e], d = dst[e];
#pragma unroll
    for (int j = 0; j < 2; ++j) {
        int c = lane + 32 * j;
        atomicAdd(&agg[d * HID + c], h2[s * HID + c]);
    }
}

__global__ void sage_div_kernel(float* __restrict__ agg, const float* __restrict__ deg) {
    int t = blockIdx.x * blockDim.x + threadIdx.x;
    if (t >= NNODES * HID) return;
    float d = deg[t >> 6];
    agg[t] /= (d < 1.f ? 1.f : d);
}

__global__ void pool_kernel(const float* __restrict__ h3, const int* __restrict__ batch,
                            float* __restrict__ gsum, float* __restrict__ cnt) {
    int n = blockIdx.x * (blockDim.x >> 5) + (threadIdx.x >> 5);
    int lane = threadIdx.x & 31;
    if (n >= NNODES) return;
    int b = batch[n];
    if (lane == 0) atomicAdd(&cnt[b], 1.f);
#pragma unroll
    for (int j = 0; j < 2; ++j) {
        int c = lane + 32 * j;
        atomicAdd(&gsum[b * HID + c], h3[n * HID + c]);
    }
}

__global__ void pool_div_kernel(float* __restrict__ gsum, const float* __restrict__ cnt) {
    int t = blockIdx.x * blockDim.x + threadIdx.x;
    if (t >= NGRAPH * HID) return;
    float c = cnt[t >> 6];
    gsum[t] /= (c < 1.f ? 1.f : c);
}

__global__ void heads_kernel(const float* __restrict__ t2,
                             const float* __restrict__ telW, const float* __restrict__ telb,
                             const float* __restrict__ compW, const float* __restrict__ compb,
                             const float* __restrict__ pchW, const float* __restrict__ pchb,
                             float* __restrict__ out) {
    int t = blockIdx.x * blockDim.x + threadIdx.x;
    if (t >= 3 * NGRAPH) return;
    int head = t / NGRAPH, r = t % NGRAPH;
    const float* W = head == 0 ? telW : (head == 1 ? compW : pchW);
    const float* b = head == 0 ? telb : (head == 1 ? compb : pchb);
    float acc = b[0];
    for (int k = 0; k < HID; ++k) acc += t2[r * HID + k] * W[k];
    out[head * NGRAPH + r] = acc;
}

// ---------------------------------------------------------------------------
// Host-side orchestration
// ---------------------------------------------------------------------------
static inline void launch_gemm(const float* A, const float* B, const float* bias, float* C,
                               int M, int Nc, int K, int accum, hipStream_t s) {
    int tiles = (M / 16) * (Nc / 64);
    int blocks = (tiles + 3) / 4;           // 4 waves / block (128 thr, wave32)
    if (Nc == 256 && K == 128)
        gemm_wmma_f32<256, 128><<<blocks, 128, 0, s>>>(A, B, bias, C, M, accum);
    else if (Nc == 256 && K == 64)
        gemm_wmma_f32<256, 64><<<blocks, 128, 0, s>>>(A, B, bias, C, M, accum);
    else // Nc == 64, K == 64
        gemm_wmma_f32<64, 64><<<blocks, 128, 0, s>>>(A, B, bias, C, M, accum);
}

struct GatScratch {
    float *hbig, *ae, *as_, *ad_, *mrow, *den, *sself;
    float *Mb, *Pb, *qb, *aeloop, *eamean;
};

static void run_gat_layer(const float* hin, int Kin, const float* W,
                          const float* asW, const float* adW, const float* aeW,
                          const float* leW, const float* bias,
                          const float* edge_attr, const float* edge_W, const float* edge_b,
                          const int* src, const int* dst, float* hout,
                          GatScratch& s, hipStream_t st) {
    launch_gemm(hin, W, nullptr, s.hbig, NNODES, HC, Kin, 0, st);

    computeM_kernel<<<1, 256, 0, st>>>(leW, aeW, s.Mb);
    computePq_kernel<<<1, 128, 0, st>>>(edge_W, edge_b, s.Mb, s.Pb, s.qb);
    aeloop_kernel<<<1, 32, 0, st>>>(s.eamean, s.Pb, s.qb, s.aeloop);

    asad_kernel<<<(NNODES * NHEAD + 255) / 256, 256, 0, st>>>(s.hbig, asW, adW, s.as_, s.ad_);
    edge_ae_kernel<<<(NEDGES * NHEAD + 255) / 256, 256, 0, st>>>(edge_attr, s.Pb, s.qb, s.ae);

    node_init_kernel<<<(NNODES * NHEAD + 255) / 256, 256, 0, st>>>(s.as_, s.ad_, s.aeloop,
                                                                   s.sself, s.mrow);
    edge_max_kernel<<<(NEDGES + 255) / 256, 256, 0, st>>>(src, dst, s.as_, s.ad_, s.ae, s.mrow);
    den_init_kernel<<<(NNODES * NHEAD + 255) / 256, 256, 0, st>>>(s.sself, s.mrow, s.den);
    edge_den_kernel<<<(NEDGES + 255) / 256, 256, 0, st>>>(src, dst, s.as_, s.ad_, s.ae,
                                                          s.mrow, s.den);
    gat_self_agg_kernel<<<(NNODES + 7) / 8, 256, 0, st>>>(s.sself, s.mrow, s.den, s.hbig, hout);
    edge_agg_kernel<<<(NEDGES + 7) / 8, 256, 0, st>>>(src, dst, s.as_, s.ad_, s.ae,
                                                      s.mrow, s.den, s.hbig, hout);
    bias_elu_kernel<<<(NNODES * HID + 255) / 256, 256, 0, st>>>(hout, bias);
}

extern "C" void kernel_launch(void* const* d_in, const int* in_sizes, int n_in,
                              void* d_out, int out_size, void* d_ws, size_t ws_size,
                              hipStream_t stream) {
    const float* x        = (const float*)d_in[0];
    const float* edge_attr= (const float*)d_in[1];
    const float* node_emb = (const float*)d_in[2];
    const float* edge_W   = (const float*)d_in[3];
    const float* edge_b   = (const float*)d_in[4];
    const float* W1       = (const float*)d_in[5];
    const float* as1      = (const float*)d_in[6];
    const float* ad1      = (const float*)d_in[7];
    const float* ae1      = (const float*)d_in[8];
    const float* leW1     = (const float*)d_in[9];
    const float* b1       = (const float*)d_in[10];
    const float* W2       = (const float*)d_in[11];
    const float* as2      = (const float*)d_in[12];
    const float* ad2      = (const float*)d_in[13];
    const float* ae2      = (const float*)d_in[14];
    const float* leW2     = (const float*)d_in[15];
    const float* b2       = (const float*)d_in[16];
    const float* sage_Wl  = (const float*)d_in[17];
    const float* sage_bl  = (const float*)d_in[18];
    const float* sage_Wr  = (const float*)d_in[19];
    const float* lin1_W   = (const float*)d_in[20];
    const float* lin1_b   = (const float*)d_in[21];
    const float* lin2_W   = (const float*)d_in[22];
    const float* lin2_b   = (const float*)d_in[23];
    const float* tel_W    = (const float*)d_in[24];
    const float* tel_b    = (const float*)d_in[25];
    const float* comp_W   = (const float*)d_in[26];
    const float* comp_b   = (const float*)d_in[27];
    const float* pch_W    = (const float*)d_in[28];
    const float* pch_b    = (const float*)d_in[29];
    const int*   edge_idx = (const int*)d_in[30];
    const int*   batch    = (const int*)d_in[31];
    const int*   node_type= (const int*)d_in[32];
    (void)in_sizes; (void)n_in; (void)out_size; (void)ws_size;

    const int* src = edge_idx;            // edge_index[0]
    const int* dst = edge_idx + NEDGES;   // edge_index[1]

    // workspace carve-up (floats); total ~87 MB
    float* p = (float*)d_ws;
    float* h0    = p; p += (size_t)NNODES * CIN1;
    float* hbig  = p; p += (size_t)NNODES * HC;
    float* ae    = p; p += (size_t)NEDGES * NHEAD;
    float* as_   = p; p += (size_t)NNODES * NHEAD;
    float* ad_   = p; p += (size_t)NNODES * NHEAD;
    float* mrow  = p; p += (size_t)NNODES * NHEAD;
    float* den   = p; p += (size_t)NNODES * NHEAD;
    float* sself = p; p += (size_t)NNODES * NHEAD;
    float* h1    = p; p += (size_t)NNODES * HID;
    float* h2    = p; p += (size_t)NNODES * HID;
    float* agg   = p; p += (size_t)NNODES * HID;
    float* h3    = p; p += (size_t)NNODES * HID;
    float* deg   = p; p += NNODES;
    float* Mb    = p; p += 256;
    float* Pb    = p; p += 64;
    float* qb    = p; p += 4;
    float* eamean= p; p += 16;
    float* aeloop= p; p += 4;
    float* gsum  = p; p += NGRAPH * HID;
    float* cnt   = p; p += NGRAPH;
    float* t1    = p; p += NGRAPH * HID;
    float* t2    = p; p += NGRAPH * HID;

    GatScratch gs{hbig, ae, as_, ad_, mrow, den, sself, Mb, Pb, qb, aeloop, eamean};

    // h0 = concat(x, node_emb[node_type]); edge_attr column means (shared by layers)
    build_h0_kernel<<<(NNODES * CIN1 + 255) / 256, 256, 0, stream>>>(x, node_emb, node_type, h0);
    colmean16_kernel<<<FEDIM, 256, 0, stream>>>(edge_attr, eamean);

    // GAT layer 1 and 2
    run_gat_layer(h0, CIN1, W1, as1, ad1, ae1, leW1, b1, edge_attr, edge_W, edge_b,
                  src, dst, h1, gs, stream);
    run_gat_layer(h1, HID,  W2, as2, ad2, ae2, leW2, b2, edge_attr, edge_W, edge_b,
                  src, dst, h2, gs, stream);

    // SAGE: mean-aggregate h2 over original edges + root transform
    hipMemsetAsync(deg, 0, NNODES * sizeof(float), stream);
    hipMemsetAsync(agg, 0, (size_t)NNODES * HID * sizeof(float), stream);
    sage_deg_kernel<<<(NEDGES + 255) / 256, 256, 0, stream>>>(dst, deg);
    sage_agg_kernel<<<(NEDGES + 7) / 8, 256, 0, stream>>>(src, dst, h2, agg);
    sage_div_kernel<<<(NNODES * HID + 255) / 256, 256, 0, stream>>>(agg, deg);
    launch_gemm(agg, sage_Wl, sage_bl, h3, NNODES, HID, HID, 0, stream);
    launch_gemm(h2,  sage_Wr, nullptr, h3, NNODES, HID, HID, 1, stream);
    elu_kernel<<<(NNODES * HID + 255) / 256, 256, 0, stream>>>(h3, NNODES * HID);

    // global mean pool per graph
    hipMemsetAsync(gsum, 0, (NGRAPH * HID + NGRAPH) * sizeof(float), stream); // gsum + cnt
    pool_kernel<<<(NNODES + 7) / 8, 256, 0, stream>>>(h3, batch, gsum, cnt);
    pool_div_kernel<<<(NGRAPH * HID + 255) / 256, 256, 0, stream>>>(gsum, cnt);

    // MLP head: t1 = elu(g@lin1+b), t2 = t1@lin2+b, then three 64->1 heads
    launch_gemm(gsum, lin1_W, lin1_b, t1, NGRAPH, HID, HID, 0, stream);
    elu_kernel<<<(NGRAPH * HID + 255) / 256, 256, 0, stream>>>(t1, NGRAPH * HID);
    launch_gemm(t1, lin2_W, lin2_b, t2, NGRAPH, HID, HID, 0, stream);
    heads_kernel<<<1, 256, 0, stream>>>(t2, tel_W, tel_b, comp_W, comp_b, pch_W, pch_b,
                                        (float*)d_out);
}